// ClusterLoss_73675868995717
// MI455X (gfx1250) — compile-verified
//
#include <hip/hip_runtime.h>

typedef float v2f __attribute__((ext_vector_type(2)));
typedef float v4f __attribute__((ext_vector_type(4)));
typedef float v8f __attribute__((ext_vector_type(8)));

#define NB  1024   // blocks in streaming pass == number of partials
#define TPB 256    // 8 wave32s per block

// ---------------------------------------------------------------------------
// Pass 1: streaming gather + squared-distance accumulation.
// Each block handles rows r = blockIdx.x, blockIdx.x+NB, ...
// 256 threads * float4 = 1024 floats = exactly one row per sweep (D=1024);
// inner loop kept general for other D.
// ---------------------------------------------------------------------------
__global__ void __launch_bounds__(TPB)
cluster_loss_partial(const float* __restrict__ X,
                     const int*   __restrict__ cid,
                     const float* __restrict__ C,
                     float*       __restrict__ partial,
                     int B, int D)
{
    const int tid       = threadIdx.x;
    const int vecPerRow = D >> 2;

    v4f acc; acc.x = 0.f; acc.y = 0.f; acc.z = 0.f; acc.w = 0.f;

    for (int r = blockIdx.x; r < B; r += gridDim.x) {
        const v4f* __restrict__ xr = (const v4f*)(X + (size_t)r * (size_t)D);
        const v4f* __restrict__ cr = (const v4f*)(C + (size_t)cid[r] * (size_t)D);
        for (int c = tid; c < vecPerRow; c += TPB) {
            v4f x  = __builtin_nontemporal_load(&xr[c]);  // streamed once: NT b128
            v4f cc = cr[c];                               // L2-resident centers
            v4f d  = x - cc;
            acc += d * d;
        }
    }

    // per-thread horizontal sum
    float t = (acc.x + acc.y) + (acc.z + acc.w);

    // wave32 tree reduction (deterministic order)
    #pragma unroll
    for (int off = 16; off > 0; off >>= 1)
        t += __shfl_down(t, off, 32);

    __shared__ float smem[TPB / 32];
    if ((tid & 31) == 0) smem[tid >> 5] = t;
    __syncthreads();

    if (tid == 0) {
        float b = 0.f;
        #pragma unroll
        for (int i = 0; i < TPB / 32; ++i) b += smem[i];
        partial[blockIdx.x] = b;
    }
}

// ---------------------------------------------------------------------------
// Pass 2: one wave32. Fold NP partials into the 64 f32 slots of a 16x4 A
// fragment, then a single v_wmma_f32_16x16x4_f32 against an all-ones B
// reduces them to 16 row-sums living in D. Lane 0 holds rows 0-7 (one per
// D VGPR), lane 16 holds rows 8-15. Sum those two lanes -> grand total.
// EXEC is all-ones (32 threads, no divergence before the WMMA).
// ---------------------------------------------------------------------------
__global__ void __launch_bounds__(32)
reduce_partials_wmma(const float* __restrict__ partial,
                     float* __restrict__ out,
                     int NP, float scale)
{
    const int lane = threadIdx.x;

    float a0 = 0.f, a1 = 0.f;
    for (int i = 2 * lane;     i < NP; i += 64) a0 += partial[i];
    for (int i = 2 * lane + 1; i < NP; i += 64) a1 += partial[i];

    v2f a;    a.x = a0;  a.y = a1;      // 16x4 f32 A fragment (2 VGPRs/lane)
    v2f ones; ones.x = 1.f; ones.y = 1.f; // 4x16 all-ones B fragment
    v8f c = {};

    v8f d = __builtin_amdgcn_wmma_f32_16x16x4_f32(
        /*neg_a=*/false, a, /*neg_b=*/false, ones,
        /*c_mod=*/(short)0, c, /*reuse_a=*/false, /*reuse_b=*/false);

    float s = 0.f;
    #pragma unroll
    for (int j = 0; j < 8; ++j) s += d[j];   // column sum of this lane's rows

    // total = (rows 0-7 @ lane 0) + (rows 8-15 @ lane 16)
    float total = __shfl(s, 0, 32) + __shfl(s, 16, 32);

    if (lane == 0) out[0] = scale * total;
}

// ---------------------------------------------------------------------------
extern "C" void kernel_launch(void* const* d_in, const int* in_sizes, int n_in,
                              void* d_out, int out_size, void* d_ws, size_t ws_size,
                              hipStream_t stream)
{
    const float* X   = (const float*)d_in[0];   // [B, D] f32
    const int*   cid = (const int*)  d_in[1];   // [B]
    const float* C   = (const float*)d_in[2];   // [K, D] f32
    float* out       = (float*)d_out;           // scalar
    float* partials  = (float*)d_ws;            // NB floats (4 KB)

    const int B = in_sizes[1];
    const int D = in_sizes[0] / B;

    cluster_loss_partial<<<NB, TPB, 0, stream>>>(X, cid, C, partials, B, D);
    reduce_partials_wmma<<<1, 32, 0, stream>>>(partials, out, NB, 0.5f /* 0.5*BETA */);
}